// BrockHommes_44538810859915
// MI455X (gfx1250) — compile-verified
//
#include <hip/hip_runtime.h>

// ---------------------------------------------------------------------------
// Brock-Hommes 4-type recurrence, 100k steps.
// Dependency distance of the scan is exactly 4 (x_t uses x_{t-4..t-6}), so we
// run 4 interleaved timesteps per "super-iteration" on one wave32:
//   lane = 4*phase + comp ; phase = which of the 4 parallel timesteps,
//   comp  = which of the 4 softmax components.
// Cross-lane work uses DPP quad_perm butterflies (VALU latency, no LDS) and
// v_readlane broadcasts. Epsilon stream is prefetched ahead with
// global_prefetch_b8. Entirely latency-bound: WMMA does not apply (no matmul).
// ---------------------------------------------------------------------------

template <int CTRL>
__device__ __forceinline__ float qperm_f32(float v) {
    // DPP16 quad_perm within each group of 4 lanes.
    return __int_as_float(
        __builtin_amdgcn_mov_dpp(__float_as_int(v), CTRL,
                                 /*row_mask=*/0xF, /*bank_mask=*/0xF,
                                 /*bound_ctrl=*/true));
}

__device__ __forceinline__ float quad_sum(float v) {
    v += qperm_f32<0xB1>(v);   // swap with lane^1 : perm [1,0,3,2]
    v += qperm_f32<0x4E>(v);   // swap with lane^2 : perm [2,3,0,1]
    return v;
}

__device__ __forceinline__ float quad_max(float v) {
    v = fmaxf(v, qperm_f32<0xB1>(v));
    v = fmaxf(v, qperm_f32<0x4E>(v));
    return v;
}

__global__ __launch_bounds__(32, 1)
void brock_hommes_scan(const float* __restrict__ P,
                       const float* __restrict__ eps,
                       float* __restrict__ out,
                       int N) {
    const int lane  = (int)(threadIdx.x & 31u);
    const int comp  = lane & 3;          // softmax component 0..3
    const int phase = (lane >> 2) & 3;   // interleaved timestep 0..3

    // Scalar parameter prep (once).
    const float beta  = expf(P[0]);
    const float gi    = P[1 + comp];
    const float bi    = P[5 + comp];
    const float sigma = expf(P[9]);
    const float R     = 1.0f + expf(P[10]);
    const float invR  = 1.0f / R;
    const float L2E   = 1.4426950408889634f; // log2(e)

    // Shared history (identical in every lane): z0..z5 = x_{t-6} .. x_{t-1}.
    float z0 = 0.f, z1 = 0.f, z2 = 0.f, z3 = 0.f, z4 = 0.f, z5 = 0.f;

    const int nb = (N + 3) >> 2;         // 4 timesteps per super-iteration
    for (int i = 0; i < nb; ++i) {
        const int t = i << 2;
        const int s = t + phase;         // this lane-group's timestep

        // Stream prefetch ~4KB ahead (global_prefetch_b8); clamp in-bounds.
        int pfi = s + 1024; pfi = pfi < N ? pfi : N - 1;
        __builtin_prefetch(eps + pfi, 0, 1);

        int li = s < N ? s : N - 1;      // tail guard (N % 4 != 0 safe)
        const float eps_s = eps[li];

        // Phase-constant selects: (xm6, xm5, xm4) = (z[k], z[k+1], z[k+2]).
        const float xm6 = phase == 0 ? z0 : phase == 1 ? z1 : phase == 2 ? z2 : z3;
        const float xm5 = phase == 0 ? z1 : phase == 1 ? z2 : phase == 2 ? z3 : z4;
        const float xm4 = phase == 0 ? z2 : phase == 1 ? z3 : phase == 2 ? z4 : z5;

        // exponent_i = beta * (xm4 - R*xm5) * (g_i*xm6 + b_i - R*xm5)
        const float rx5 = R * xm5;
        const float c1  = xm4 - rx5;
        const float c2  = fmaf(gi, xm6, bi) - rx5;
        const float ex  = beta * c1 * c2;

        // Numerically-stable softmax over the 4 component lanes.
        const float m  = quad_max(ex);
        const float e  = __builtin_amdgcn_exp2f((ex - m) * L2E);  // exp(ex-m)
        const float yv = fmaf(gi, xm4, bi);                       // g*xm4 + b

        const float den = quad_sum(e);
        const float num = quad_sum(e * yv);
        const float mean = num * __builtin_amdgcn_rcpf(den);

        // x_t = (mean + eps*sigma) / R
        const float x = fmaf(eps_s, sigma, mean) * invR;

        if (comp == 0 && lane < 16 && s < N) out[s] = x;

        // Broadcast the 4 new values (groups 0..3 -> lanes 0,4,8,12) and
        // shift the shared 6-deep history by 4.
        const int   xb  = __float_as_int(x);
        const float y0f = __int_as_float(__builtin_amdgcn_readlane(xb, 0));
        const float y1f = __int_as_float(__builtin_amdgcn_readlane(xb, 4));
        const float y2f = __int_as_float(__builtin_amdgcn_readlane(xb, 8));
        const float y3f = __int_as_float(__builtin_amdgcn_readlane(xb, 12));

        z0 = z4; z1 = z5; z2 = y0f; z3 = y1f; z4 = y2f; z5 = y3f;
    }
}

extern "C" void kernel_launch(void* const* d_in, const int* in_sizes, int n_in,
                              void* d_out, int out_size, void* d_ws, size_t ws_size,
                              hipStream_t stream) {
    (void)n_in; (void)d_ws; (void)ws_size; (void)out_size;
    const float* params = (const float*)d_in[0];   // 11 floats
    const float* eps    = (const float*)d_in[1];   // N_TIMESTEPS floats
    float*       out    = (float*)d_out;           // N_TIMESTEPS floats
    const int n = in_sizes[1];

    // Inherently serial scan: one wave, latency-optimized inner loop.
    brock_hommes_scan<<<dim3(1), dim3(32), 0, stream>>>(params, eps, out, n);
}